// MultiHeadAttention_26809185861863
// MI455X (gfx1250) — compile-verified
//
#include <hip/hip_runtime.h>

#define S_LEN   2048
#define DMODEL  1024
#define NHEADS  16
#define DK      64

typedef __attribute__((ext_vector_type(2))) float v2f;
typedef __attribute__((ext_vector_type(8))) float v8f;

// ---------------------------------------------------------------------------
// fp32 WMMA GEMM with 4x4 register blocking:  C = scale * (A @ op(B)) + bias
//   A: [M,K] row-major (lda), batched by sA elements
//   TRANSB=true : B is [N,K] row-major (ldb)  -> C = A * B^T
//   TRANSB=false: B is [K,N] row-major (ldb)  -> C = A * B
// One wave computes a 64x64 C block (16 tiles of 16x16), stepping K by 4 with
// v_wmma_f32_16x16x4_f32.  Requires M,N multiples of 64; K multiple of 4.
// A-fragment layout (ISA 7.12.2, 32-bit A 16x4): lane = m + 16*(k>>1),
// vgpr = k&1  ->  each lane loads a contiguous float2 from row (m = lane&15)
// at column kbase + 2*(lane>>4).  B mirrors with n = lane&15.
// Per K-step: 8 fragment loads feed 16 WMMAs  ->  16 FLOP/byte.
// ---------------------------------------------------------------------------
template <bool TRANSB>
__global__ __launch_bounds__(256) void gemm_wmma_f32(
    const float* __restrict__ A, int lda, long long sA,
    const float* __restrict__ B, int ldb, long long sB,
    float* __restrict__ C, int ldc, long long sC,
    int M, int N, int K, float scale, const float* __restrict__ bias)
{
  const int batch = blockIdx.y;
  A += (long long)batch * sA;
  B += (long long)batch * sB;
  C += (long long)batch * sC;

  const int wave   = threadIdx.x >> 5;
  const int lane   = threadIdx.x & 31;
  const int laneLo = lane & 15;
  const int laneHi = lane >> 4;

  const int blocksN   = N >> 6;
  const int blocksAll = (M >> 6) * blocksN;
  const int blkIdx    = blockIdx.x * (blockDim.x >> 5) + wave;
  if (blkIdx >= blocksAll) return;        // wave-uniform: EXEC stays all-ones
  const int bm = blkIdx / blocksN;
  const int bn = blkIdx % blocksN;

  // Per-lane base pointers for the 4 row-fragments of A and of B.
  const float* ap = A + (long long)(bm * 64 + laneLo) * lda + 2 * laneHi;
  const float* bp = TRANSB
      ? (B + (long long)(bn * 64 + laneLo) * ldb + 2 * laneHi)   // row n of B
      : (B + (long long)(2 * laneHi) * ldb + bn * 64 + laneLo);  // col n of B

  v8f acc[16];
  {
    v8f z = {};
    #pragma unroll
    for (int i = 0; i < 16; ++i) acc[i] = z;
  }

  #pragma unroll 2
  for (int k = 0; k < K; k += 4) {
    v2f a[4], b[4];
    #pragma unroll
    for (int i = 0; i < 4; ++i)
      a[i] = *(const v2f*)(ap + (long long)(i * 16) * lda + k);
    #pragma unroll
    for (int j = 0; j < 4; ++j) {
      if (TRANSB) {
        b[j] = *(const v2f*)(bp + (long long)(j * 16) * ldb + k);
      } else {
        const float* bq = bp + (long long)k * ldb + j * 16;
        b[j].x = bq[0];
        b[j].y = bq[ldb];
      }
    }
    #pragma unroll
    for (int i = 0; i < 4; ++i)
      #pragma unroll
      for (int j = 0; j < 4; ++j)
        acc[i * 4 + j] = __builtin_amdgcn_wmma_f32_16x16x4_f32(
            false, a[i], false, b[j], (short)0, acc[i * 4 + j], false, false);
  }

  // Store 64x64 block.  C tile layout: row m = v + 8*laneHi, col n = laneLo.
  #pragma unroll
  for (int j = 0; j < 4; ++j) {
    const int col = bn * 64 + j * 16 + laneLo;
    const float bv = bias ? bias[col] : 0.0f;
    #pragma unroll
    for (int i = 0; i < 4; ++i) {
      float* crow = C + (long long)(bm * 64 + i * 16 + 8 * laneHi) * ldc + col;
      #pragma unroll
      for (int v = 0; v < 8; ++v)
        crow[(long long)v * ldc] = acc[i * 4 + j][v] * scale + bv;
    }
  }
}

// ---------------------------------------------------------------------------
// RoPE in place on Q and K (layout [S, H*DK]).  One thread per rotation pair
// (s, h, jj) with jj in [0,32): rotates (x[jj], x[jj+32]) by angle s*inv_freq.
// Pairwise update -> no in-place race.
// ---------------------------------------------------------------------------
__global__ __launch_bounds__(256) void rope_kernel(float* __restrict__ Q,
                                                   float* __restrict__ Km)
{
  const int idx = blockIdx.x * blockDim.x + threadIdx.x;
  const int total = S_LEN * NHEADS * (DK / 2);
  if (idx >= total) return;
  const int jj = idx & 31;
  const int h  = (idx >> 5) & (NHEADS - 1);
  const int s  = idx >> 9;

  // inv_freq = 10000^(-2*jj/64) = exp(-(jj/32)*ln(10000))
  const float inv = expf(-(float)jj * (9.210340371976184f / 32.0f));
  const float ang = (float)s * inv;
  const float c = cosf(ang), sn = sinf(ang);

  const long long base = (long long)s * DMODEL + h * DK + jj;
  float x1 = Q[base], x2 = Q[base + 32];
  Q[base]      = x1 * c - x2 * sn;
  Q[base + 32] = x2 * c + x1 * sn;
  x1 = Km[base]; x2 = Km[base + 32];
  Km[base]      = x1 * c - x2 * sn;
  Km[base + 32] = x2 * c + x1 * sn;
}

// ---------------------------------------------------------------------------
// Row softmax over the attention tensor: one block per row of 2048 floats,
// 256 threads x 8 values held in registers -> single global read + write.
// ---------------------------------------------------------------------------
__global__ __launch_bounds__(256) void softmax_rows(float* __restrict__ attn)
{
  __shared__ float red[8];
  const long long row = blockIdx.x;
  float* p = attn + row * (long long)S_LEN;
  const int t = threadIdx.x;

  float vals[8];
  float m = -3.4e38f;
  #pragma unroll
  for (int i = 0; i < 8; ++i) {
    vals[i] = p[t + (i << 8)];
    m = fmaxf(m, vals[i]);
  }
  #pragma unroll
  for (int off = 16; off > 0; off >>= 1)
    m = fmaxf(m, __shfl_xor(m, off, 32));
  if ((t & 31) == 0) red[t >> 5] = m;
  __syncthreads();
  m = red[0];
  #pragma unroll
  for (int i = 1; i < 8; ++i) m = fmaxf(m, red[i]);
  __syncthreads();

  float s = 0.0f;
  #pragma unroll
  for (int i = 0; i < 8; ++i) {
    vals[i] = __expf(vals[i] - m);
    s += vals[i];
  }
  #pragma unroll
  for (int off = 16; off > 0; off >>= 1)
    s += __shfl_xor(s, off, 32);
  if ((t & 31) == 0) red[t >> 5] = s;
  __syncthreads();
  s = red[0];
  #pragma unroll
  for (int i = 1; i < 8; ++i) s += red[i];

  const float invs = 1.0f / s;
  #pragma unroll
  for (int i = 0; i < 8; ++i) p[t + (i << 8)] = vals[i] * invs;
}

// ---------------------------------------------------------------------------
extern "C" void kernel_launch(void* const* d_in, const int* in_sizes, int n_in,
                              void* d_out, int out_size, void* d_ws, size_t ws_size,
                              hipStream_t stream)
{
  const float* q_in = (const float*)d_in[0];
  const float* k_in = (const float*)d_in[1];
  const float* v_in = (const float*)d_in[2];
  const float* w_q  = (const float*)d_in[3];
  const float* w_k  = (const float*)d_in[4];
  const float* w_v  = (const float*)d_in[5];
  const float* w_o  = (const float*)d_in[6];
  const float* b_o  = (const float*)d_in[7];

  float* out  = (float*)d_out;                         // [2048,1024]
  float* attn = out + (size_t)S_LEN * DMODEL;          // [16,2048,2048]

  float* Q   = (float*)d_ws;                           // [2048,1024]  8 MB
  float* Kp  = Q + (size_t)S_LEN * DMODEL;             // [2048,1024]  8 MB
  float* V   = Kp + (size_t)S_LEN * DMODEL;            // [2048,1024]  8 MB
  float* ctx = Q;   // Q is dead after the score GEMM -> reuse for ctx

  const dim3 blk(256);

  // 1) projections: Q/K/V = X @ W^T   (M=2048, N=1024, K=1024)
  const int blocksP = (S_LEN / 64) * (DMODEL / 64);    // 512 64x64 blocks
  const dim3 gP((blocksP + 7) / 8, 1);
  gemm_wmma_f32<true><<<gP, blk, 0, stream>>>(
      q_in, DMODEL, 0, w_q, DMODEL, 0, Q, DMODEL, 0,
      S_LEN, DMODEL, DMODEL, 1.0f, nullptr);
  gemm_wmma_f32<true><<<gP, blk, 0, stream>>>(
      k_in, DMODEL, 0, w_k, DMODEL, 0, Kp, DMODEL, 0,
      S_LEN, DMODEL, DMODEL, 1.0f, nullptr);
  gemm_wmma_f32<true><<<gP, blk, 0, stream>>>(
      v_in, DMODEL, 0, w_v, DMODEL, 0, V, DMODEL, 0,
      S_LEN, DMODEL, DMODEL, 1.0f, nullptr);

  // 2) RoPE on Q and K in place
  const int pairs = S_LEN * NHEADS * (DK / 2);         // 1M threads
  rope_kernel<<<(pairs + 255) / 256, blk, 0, stream>>>(Q, Kp);

  // 3) scores = (Q_h @ K_h^T) / 8  -> attn region  (batched over heads)
  const int blocksS = (S_LEN / 64) * (S_LEN / 64);     // 1024 blocks/head
  const dim3 gS((blocksS + 7) / 8, NHEADS);
  gemm_wmma_f32<true><<<gS, blk, 0, stream>>>(
      Q,  DMODEL, (long long)DK,
      Kp, DMODEL, (long long)DK,
      attn, S_LEN, (long long)S_LEN * S_LEN,
      S_LEN, S_LEN, DK, 0.125f, nullptr);

  // 4) softmax over each of the 16*2048 rows
  softmax_rows<<<NHEADS * S_LEN, blk, 0, stream>>>(attn);

  // 5) ctx = attn_h @ V_h   (NN, batched over heads)  M=2048,N=64,K=2048
  const int blocksC = (S_LEN / 64) * (DK / 64);        // 32 blocks/head
  const dim3 gC((blocksC + 7) / 8, NHEADS);
  gemm_wmma_f32<false><<<gC, blk, 0, stream>>>(
      attn, S_LEN, (long long)S_LEN * S_LEN,
      V,    DMODEL, (long long)DK,
      ctx,  DMODEL, (long long)DK,
      S_LEN, DK, S_LEN, 1.0f, nullptr);

  // 6) out = ctx @ w_o^T + b_o
  gemm_wmma_f32<true><<<gP, blk, 0, stream>>>(
      ctx, DMODEL, 0, w_o, DMODEL, 0, out, DMODEL, 0,
      S_LEN, DMODEL, DMODEL, 1.0f, b_o);
}